// DecodePredictions_51840255262825
// MI455X (gfx1250) — compile-verified
//
#include <hip/hip_runtime.h>
#include <hip/hip_bf16.h>
#include <stdint.h>

#pragma clang fp contract(off)

// ---------------- problem constants (match reference) ----------------
namespace {
constexpr int NB   = 16;      // batch
constexpr int NA   = 110484;  // anchors (768x768, strides 8..128, 9 per cell)
constexpr int NC   = 80;      // classes
constexpr int NK   = 300;     // per-class pre-NMS candidates
constexpr int NDET = 100;     // max detections
constexpr int NF   = NC * NK; // 24000 flat candidates per image
constexpr int SLOTS = (NF + 1023) / 1024; // 24 per-thread slots in final top-k
}

// gfx1250 async global->LDS path (guarded; falls back to plain loads)
#if defined(__HIP_DEVICE_COMPILE__) && __has_builtin(__builtin_amdgcn_global_load_async_to_lds_b128)
#define HAVE_ASYNC_LDS 1
typedef int av4i __attribute__((vector_size(16)));           // 4 x i32 vector
typedef __attribute__((address_space(1))) av4i* g_v4i_ptr;   // global pointer
typedef __attribute__((address_space(3))) av4i* l_v4i_ptr;   // LDS pointer
#else
#define HAVE_ASYNC_LDS 0
#endif

// order-preserving float -> uint32 key (larger key == larger float)
__device__ __forceinline__ unsigned ukey(float f) {
  unsigned b = __float_as_uint(f);
  return b ^ ((unsigned)((int)b >> 31) | 0x80000000u);
}
__device__ __forceinline__ float key2f(unsigned u) {
  unsigned b = (u & 0x80000000u) ? (u ^ 0x80000000u) : ~u;
  return __uint_as_float(b);
}

// ---------------- K0: decode anchors + boxes -> corner boxes [B,A,4] ----------------
__global__ void k_decode(const float* __restrict__ pred, float* __restrict__ boxes) {
  __shared__ float sdw[45], sdh[45]; // 5 levels x 9 anchor dims, f32-rounded from double
  const int t = threadIdx.x;
  if (t < 45) {
    int lvl = t / 9, k = t % 9;
    double side = (double)(32 << lvl);          // 32,64,128,256,512
    double area = side * side;
    int ri = k / 3, si = k % 3;                 // ratio-major, then scale
    double r = (ri == 0) ? 0.5 : ((ri == 1) ? 1.0 : 2.0);
    double s = (si == 0) ? 1.0 : pow(2.0, (double)si / 3.0);
    double ah = sqrt(area / r);
    double aw = area / ah;
    sdw[t] = (float)(s * aw);
    sdh[t] = (float)(s * ah);
  }
  __syncthreads();
  const long long gid = (long long)blockIdx.x * blockDim.x + t; // b*NA + a
  if (gid >= (long long)NB * NA) return;
  const int a = (int)(gid % NA);
  int lvl, base;
  if      (a < 82944)  { lvl = 0; base = 0; }
  else if (a < 103680) { lvl = 1; base = 82944; }
  else if (a < 108864) { lvl = 2; base = 103680; }
  else if (a < 110160) { lvl = 3; base = 108864; }
  else                 { lvl = 4; base = 110160; }
  const int fw = 96 >> lvl;
  const float strideF = (float)(8 << lvl);
  const int idx  = a - base;
  const int k    = idx % 9;
  const int cell = idx / 9;
  const int gx = cell % fw;
  const int gy = cell / fw;
  const float acx = ((float)gx + 0.5f) * strideF;
  const float acy = ((float)gy + 0.5f) * strideF;
  const float aw = sdw[lvl * 9 + k];
  const float ah = sdh[lvl * 9 + k];
  const float4 p = *(const float4*)(pred + (size_t)gid * 84); // 84-float rows, 16B aligned
  float bx = p.x * 0.1f, by = p.y * 0.1f, bw = p.z * 0.2f, bh = p.w * 0.2f;
  float cx = bx * aw + acx;
  float cy = by * ah + acy;
  float w  = expf(bw) * aw;
  float h  = expf(bh) * ah;
  float4 o;
  o.x = cx - w * 0.5f;
  o.y = cy - h * 0.5f;
  o.z = cx + w * 0.5f;
  o.w = cy + h * 0.5f;
  *(float4*)(boxes + (size_t)gid * 4) = o;
}

// ------- K1: per-batch sigmoid + transpose: pred[b,:,4:84] -> scoresT[C,A] (L2-resident) -------
__global__ void k_sig_transpose(const float* __restrict__ predB, float* __restrict__ scoresT) {
  __shared__ __align__(16) float tile[32][33];
  const int a0 = blockIdx.x * 32;
  const int c0 = blockIdx.y * 32;
  const int tx = threadIdx.x, ty = threadIdx.y;
  const int a = a0 + ty, c = c0 + tx;
  if (a < NA && c < NC) {
    float x = predB[(size_t)a * 84 + 4 + c]; // coalesced along class dim
    tile[ty][tx] = 1.0f / (1.0f + expf(-x));
  }
  __syncthreads();
  const int wa = a0 + tx, wc = c0 + ty;
  if (wa < NA && wc < NC)
    scoresT[(size_t)wc * NA + wa] = tile[tx][ty]; // coalesced along anchor dim
}

// ------- K2: exact stable top-300 per class via 4x8-bit radix refinement + bitonic sort -------
__global__ __launch_bounds__(1024)
void k_topk(const float* __restrict__ scoresT, float* __restrict__ topS, int* __restrict__ topA) {
  const int c = blockIdx.x;
  const int t = threadIdx.x;
  const float* row = scoresT + (size_t)c * NA;
  __shared__ unsigned hist[256];
  __shared__ unsigned selByte, selG;
  __shared__ unsigned cntG, cntE;
  __shared__ unsigned long long skey[1024];
  __shared__ unsigned long long ekey[512];

  unsigned prefix = 0, G = 0;
  for (int r = 0; r < 4; ++r) {
    const int sh = 24 - 8 * r;
    if (t < 256) hist[t] = 0;
    __syncthreads();
    for (int a = t; a < NA; a += 1024) {
      unsigned u = ukey(row[a]);
      if (r == 0 || (u >> (sh + 8)) == prefix)
        atomicAdd(&hist[(u >> sh) & 0xFFu], 1u);
    }
    __syncthreads();
    if (t == 0) {
      unsigned cum = 0, chosen = 0, Gn = G;
      for (int bin = 255; bin >= 0; --bin) {
        unsigned cc = cum + hist[bin];
        if (G + cc >= (unsigned)NK) { chosen = (unsigned)bin; Gn = G + cum; break; }
        cum = cc;
      }
      selByte = chosen; selG = Gn;
    }
    __syncthreads();
    prefix = (prefix << 8) | selByte;
    G = selG;
    __syncthreads();
  }
  const unsigned T = prefix; // exact key of the NK-th largest; G = count(key > T) < NK

  if (t == 0) { cntG = 0; cntE = 0; }
  __syncthreads();
  for (int a = t; a < NA; a += 1024) {
    __builtin_prefetch(row + a + 4096, 0, 0); // -> global_prefetch_b8
    unsigned u = ukey(row[a]);
    if (u > T) {
      unsigned p = atomicAdd(&cntG, 1u);
      skey[p] = ((unsigned long long)u << 32) | (unsigned)(~a);
    } else if (u == T) {
      unsigned p = atomicAdd(&cntE, 1u);
      if (p < 512u) ekey[p] = ((unsigned long long)u << 32) | (unsigned)(~a);
    }
  }
  __syncthreads();
  const unsigned nG = cntG;
  const unsigned nE = cntE < 512u ? cntE : 512u;
  if ((unsigned)t >= nG) {
    unsigned e = (unsigned)t - nG;
    skey[t] = (e < nE) ? ekey[e] : 0ULL;
  }
  __syncthreads();
  // bitonic sort, descending; key = (score_key, ~index) => value desc, index asc (stable)
  for (unsigned kk = 2; kk <= 1024; kk <<= 1) {
    for (unsigned j = kk >> 1; j > 0; j >>= 1) {
      const unsigned i = (unsigned)t;
      const unsigned ixj = i ^ j;
      if (ixj > i) {
        unsigned long long va = skey[i], vb = skey[ixj];
        bool descBlock = ((i & kk) == 0);
        if (descBlock ? (va < vb) : (va > vb)) { skey[i] = vb; skey[ixj] = va; }
      }
      __syncthreads();
    }
  }
  if (t < NK) {
    unsigned long long kv = skey[t];
    topS[(size_t)c * NK + t] = key2f((unsigned)(kv >> 32));
    topA[(size_t)c * NK + t] = (int)~(unsigned)kv;
  }
}

// ------- K3: per-(b,c) greedy NMS over 300 sorted candidates -------
__global__ __launch_bounds__(320)
void k_nms(const float* __restrict__ topS, const int* __restrict__ topA,
           const float* __restrict__ boxes, float* __restrict__ flatS) {
  const int bc = blockIdx.x;
  const int c = bc % NC;
  const int b = bc / NC;
  const int t = threadIdx.x;
  __shared__ __align__(16) float sbox[NK * 4];   // 16B-aligned: async B128 LDS writes
  __shared__ float bx1[NK], by1[NK], bx2[NK], by2[NK], barea[NK], bscore[NK];
  __shared__ unsigned char sup[NK], keepA[NK];
  __shared__ int curKeep;

  const size_t base = ((size_t)b * NC + c) * NK;
  if (t < NK) {
    const int anchor = topA[base + t];
    const float* gb = boxes + ((size_t)b * NA + anchor) * 4;
#if HAVE_ASYNC_LDS
    // per-lane gather: 16B global -> 16B LDS via the async tensor/LDS engine
    g_v4i_ptr gsrc = (g_v4i_ptr)(uintptr_t)gb;
    l_v4i_ptr ldst = (l_v4i_ptr)(uint32_t)(uintptr_t)&sbox[4 * t];
    __builtin_amdgcn_global_load_async_to_lds_b128(gsrc, ldst, 0, 0);
#else
    const float4 v = *(const float4*)gb;
    sbox[4 * t + 0] = v.x; sbox[4 * t + 1] = v.y;
    sbox[4 * t + 2] = v.z; sbox[4 * t + 3] = v.w;
#endif
  }
#if HAVE_ASYNC_LDS
#if __has_builtin(__builtin_amdgcn_s_wait_asynccnt)
  __builtin_amdgcn_s_wait_asynccnt(0);
#else
  asm volatile("s_wait_asynccnt 0" ::: "memory");
#endif
#endif
  __syncthreads();
  if (t < NK) {
    float x1 = sbox[4 * t + 0], y1 = sbox[4 * t + 1];
    float x2 = sbox[4 * t + 2], y2 = sbox[4 * t + 3];
    bx1[t] = x1; by1[t] = y1; bx2[t] = x2; by2[t] = y2;
    barea[t] = (x2 - x1) * (y2 - y1);
    bscore[t] = topS[base + t];
    sup[t] = 0; keepA[t] = 0;
  }
  __syncthreads();
  for (int i = 0; i < NK; ++i) {
    if (t == 0) {
      int kf = (bscore[i] > 0.05f) && !sup[i];
      keepA[i] = (unsigned char)kf;
      curKeep = kf;
    }
    __syncthreads();
    if (curKeep && t < NK && t != i && !sup[t]) {
      float lx = fmaxf(bx1[t], bx1[i]);
      float ly = fmaxf(by1[t], by1[i]);
      float rx = fminf(bx2[t], bx2[i]);
      float ry = fminf(by2[t], by2[i]);
      float wv = rx - lx; wv = wv > 0.0f ? wv : 0.0f;
      float hv = ry - ly; hv = hv > 0.0f ? hv : 0.0f;
      float inter = wv * hv;
      float iou = inter / (barea[t] + barea[i] - inter + 1e-8f);
      if (iou > 0.5f) sup[t] = 1;
    }
    __syncthreads();
  }
  if (t < NK)
    flatS[(size_t)b * NF + (size_t)c * NK + t] = keepA[t] ? bscore[t] : -1.0f;
}

// ------- K4: global stable top-100 per image (iterative block-max over register keys) -------
__global__ __launch_bounds__(1024)
void k_final(const float* __restrict__ flatS, const int* __restrict__ topA,
             const float* __restrict__ boxes, float* __restrict__ outB,
             float* __restrict__ outS, float* __restrict__ outC) {
  const int b = blockIdx.x;
  const int t = threadIdx.x;
  const float* fs = flatS + (size_t)b * NF;
  unsigned long long lk[SLOTS];
#pragma unroll
  for (int s = 0; s < SLOTS; ++s) {
    const int idx = t + s * 1024;
    if (idx < NF) {
      unsigned u = ukey(fs[idx]);
      lk[s] = ((unsigned long long)u << 32) | (unsigned)(~idx); // value desc, index asc
    } else {
      lk[s] = 0ULL;
    }
  }
  __shared__ unsigned long long red[1024];
  for (int it = 0; it < NDET; ++it) {
    unsigned long long m = 0;
#pragma unroll
    for (int s = 0; s < SLOTS; ++s) m = lk[s] > m ? lk[s] : m;
    red[t] = m;
    __syncthreads();
    for (int off = 512; off > 0; off >>= 1) {
      if (t < off) { if (red[t + off] > red[t]) red[t] = red[t + off]; }
      __syncthreads();
    }
    const unsigned long long best = red[0];
#pragma unroll
    for (int s = 0; s < SLOTS; ++s) if (lk[s] == best) lk[s] = 0ULL; // keys unique
    if (t == 0) {
      const unsigned u = (unsigned)(best >> 32);
      const int fidx = (int)~(unsigned)best;
      const float sc = key2f(u);
      outS[b * NDET + it] = sc;
      const int cls = fidx / NK;
      const int kk  = fidx - cls * NK;
      outC[b * NDET + it] = (sc > 0.0f) ? (float)cls : -1.0f;
      const int anchor = topA[((size_t)b * NC + cls) * NK + kk];
      const float* gb = boxes + ((size_t)b * NA + anchor) * 4;
      float* ob = outB + ((size_t)b * NDET + it) * 4;
      ob[0] = gb[0]; ob[1] = gb[1]; ob[2] = gb[2]; ob[3] = gb[3];
    }
    __syncthreads();
  }
}

// ---------------- host launch ----------------
extern "C" void kernel_launch(void* const* d_in, const int* in_sizes, int n_in,
                              void* d_out, int out_size, void* d_ws, size_t ws_size,
                              hipStream_t stream) {
  (void)in_sizes; (void)n_in; (void)out_size; (void)ws_size;
  const float* pred = (const float*)d_in[1]; // d_in[0] = images (only shape 768x768 used)

  char* wsb = (char*)d_ws;
  const size_t offBoxes = 0;
  const size_t offScoT  = offBoxes + (size_t)NB * NA * 4 * sizeof(float); // 28.3 MB
  const size_t offTopS  = offScoT  + (size_t)NC * NA * sizeof(float);     // +35.4 MB
  const size_t offTopA  = offTopS  + (size_t)NB * NC * NK * sizeof(float);
  const size_t offFlat  = offTopA  + (size_t)NB * NC * NK * sizeof(int);
  float* boxes  = (float*)(wsb + offBoxes);
  float* scoT   = (float*)(wsb + offScoT);
  float* topS   = (float*)(wsb + offTopS);
  int*   topA   = (int*)  (wsb + offTopA);
  float* flatS  = (float*)(wsb + offFlat);

  { // K0: decode all boxes
    const long long total = (long long)NB * NA;
    const int blocks = (int)((total + 255) / 256);
    k_decode<<<blocks, 256, 0, stream>>>(pred, boxes);
  }
  // per-batch rounds: the 35 MB score plane stays resident in the 192 MB L2
  for (int b = 0; b < NB; ++b) {
    dim3 g1((NA + 31) / 32, (NC + 31) / 32);
    k_sig_transpose<<<g1, dim3(32, 32), 0, stream>>>(pred + (size_t)b * NA * 84, scoT);
    k_topk<<<NC, 1024, 0, stream>>>(scoT, topS + (size_t)b * NC * NK, topA + (size_t)b * NC * NK);
  }
  k_nms<<<NB * NC, 320, 0, stream>>>(topS, topA, boxes, flatS);

  float* outB = (float*)d_out;
  float* outS = outB + (size_t)NB * NDET * 4;
  float* outC = outS + (size_t)NB * NDET;
  k_final<<<NB, 1024, 0, stream>>>(flatS, topA, boxes, outB, outS, outC);
}